// SpectralMapDecomposition_9955734192129
// MI455X (gfx1250) — compile-verified
//
#include <hip/hip_runtime.h>

typedef __attribute__((ext_vector_type(2))) float v2f;
typedef __attribute__((ext_vector_type(8))) float v8f;

#define BATCH 16
#define HH    128
#define WW    128
#define EE    6
#define ROWS  (BATCH * HH)   // 2048
#define KD    256            // [xr | xi]
#define ND    256            // [F_re | F_im]
#define NCH   (2 * WW + EE)  // 262
#define PI_F  3.14159265358979f

// ---------------------------------------------------------------------------
// Stage 1: F(2048x256) = [xr|xi](2048x256) @ T(256x256) via v_wmma_f32_16x16x4_f32
// T[k][n]: n<128 (F_re, i=n):   k<128 ?  cos(2pi*i*k/W) : sin(2pi*i*(k-128)/W)
//          n>=128 (F_im, i=n-128): k<128 ? -sin(2pi*i*k/W) : cos(2pi*i*(k-128)/W)
// One wave computes one 16x16 tile; 8 waves/block share the B tile in LDS.
// LDS layout pairs (k, k+1) adjacently so a B fragment is a single b64 load.
// ---------------------------------------------------------------------------
__global__ void __launch_bounds__(256)
dft_rows_wmma(const float* __restrict__ recon, float* __restrict__ Fout)
{
    __shared__ float Bs[KD * 16];   // 16 KB, paired-k layout

    const int tid   = threadIdx.x;
    const int ncol0 = blockIdx.y * 16;

    // Build twiddle B-tile (256 x 16) in LDS
    for (int idx = tid; idx < KD * 16; idx += 256) {
        int k  = idx >> 4;
        int n0 = idx & 15;
        int n  = ncol0 + n0;
        int i  = n & 127;
        int kw = k & 127;
        float theta = (2.0f * PI_F / (float)WW) * (float)((i * kw) & 127);
        float s, c;
        __sincosf(theta, &s, &c);
        float v;
        if (n < 128) v = (k < 128) ?  c : s;
        else         v = (k < 128) ? -s : c;
        // paired layout: element (k, n0) -> [(k>>1)*32 + n0*2 + (k&1)]
        Bs[((k >> 1) << 5) + (n0 << 1) + (k & 1)] = v;
    }
    __syncthreads();

    const int lane  = tid & 31;
    const int wv    = tid >> 5;
    const int tm0   = blockIdx.x * 128 + wv * 16;   // tile row base
    const int m     = tm0 + (lane & 15);            // this lane's A row
    const int b     = m >> 7;
    const int h     = m & 127;
    const int khalf = (lane < 16) ? 0 : 2;          // A frag: upper half-wave holds K+2,K+3
    const int n0    = lane & 15;

    const float* baseR = recon + (size_t)((b * 2    ) * HH + h) * WW;  // xr row
    const float* baseI = recon + (size_t)((b * 2 + 1) * HH + h) * WW;  // xi row

    v8f acc = {0.f, 0.f, 0.f, 0.f, 0.f, 0.f, 0.f, 0.f};

#pragma unroll 4
    for (int k = 0; k < KD; k += 4) {
        int k0 = k + khalf;                 // even; (k0,k0+1) same half since k%4==0
        const float* ap = (k0 < 128) ? (baseR + k0) : (baseI + (k0 - 128));
        v2f a = *(const v2f*)ap;
        v2f bf = *(const v2f*)&Bs[((k0 >> 1) << 5) + (n0 << 1)];
        acc = __builtin_amdgcn_wmma_f32_16x16x4_f32(
            /*neg_a=*/false, a, /*neg_b=*/false, bf,
            /*c_mod=*/(short)0, acc, /*reuse_a=*/false, /*reuse_b=*/false);
    }

    // C/D layout: VGPR r -> M = r (lanes 0-15) or r+8 (lanes 16-31), N = lane&15
    const int rbase = tm0 + ((lane < 16) ? 0 : 8);
    const int col   = ncol0 + (lane & 15);
#pragma unroll
    for (int r = 0; r < 8; ++r) {
        Fout[(size_t)(rbase + r) * ND + col] = acc[r];
    }
}

// ---------------------------------------------------------------------------
// Stage 2: stream 275 MB of output (store-BW bound). One thread per float4.
// out[b,c,h,w]:
//   c<128:  (1/W)*( Fre*cos - Fim*sin ) + mask[b,c]
//   c<256:  i=c-128, (1/W)*( Fim*cos + Fre*sin ) + mask[b,i]
//   c>=256: mask_embedding[b, c-256, h, w]
// ---------------------------------------------------------------------------
__global__ void __launch_bounds__(256)
expand_output(const float* __restrict__ Fws, const float* __restrict__ memb,
              const float* __restrict__ mask, float* __restrict__ out, int total4)
{
    const float scale = 1.0f / 128.0f;
    int idx = blockIdx.x * 256 + threadIdx.x;
    if (idx >= total4) return;

    int w4 = idx & 31;           // 32 float4 per row of 128
    int t  = idx >> 5;
    int h  = t & 127;  t >>= 7;
    int c  = t % NCH;
    int b  = t / NCH;
    int w  = w4 << 2;

    float4 res;
    if (c < 256) {
        int   i    = c & 127;
        int   row  = (b << 7) + h;
        float fre  = Fws[(size_t)row * ND + i];
        float fim  = Fws[(size_t)row * ND + 128 + i];
        float mv   = mask[(b << 7) + i];
        bool  imag = (c >= 128);
        float* rp  = (float*)&res;
#pragma unroll
        for (int j = 0; j < 4; ++j) {
            int ww = w + j;
            float theta = (2.0f * PI_F / 128.0f) * (float)((i * ww) & 127);
            float s, cth;
            __sincosf(theta, &s, &cth);
            float re = scale * (fre * cth - fim * s);
            float im = scale * (fim * cth + fre * s);
            rp[j] = (imag ? im : re) + mv;
        }
    } else {
        int e = c - 256;
        res = *(const float4*)&memb[((size_t)((b * EE + e) << 7) + h) * WW + w];
    }
    *(float4*)&out[((size_t)(b * NCH + c) * HH + h) * WW + w] = res;
}

// ---------------------------------------------------------------------------
extern "C" void kernel_launch(void* const* d_in, const int* in_sizes, int n_in,
                              void* d_out, int out_size, void* d_ws, size_t ws_size,
                              hipStream_t stream)
{
    const float* recon = (const float*)d_in[0];  // [16,2,128,128]
    const float* memb  = (const float*)d_in[1];  // [16,6,128,128]
    const float* mask  = (const float*)d_in[2];  // [16,1,1,128]
    float*       out   = (float*)d_out;          // [16,262,128,128]
    float*       Fws   = (float*)d_ws;           // 2048 x 256 f32 = 2 MB scratch

    // Stage 1: row-DFT GEMM. Grid: 16 M-tiles (128 rows each) x 16 N-tiles.
    dim3 g1(ROWS / 128, ND / 16);
    dft_rows_wmma<<<g1, 256, 0, stream>>>(recon, Fws);

    // Stage 2: expansion, one thread per float4 of output.
    int total4 = BATCH * NCH * HH * (WW / 4);
    int blocks = (total4 + 255) / 256;
    expand_output<<<blocks, 256, 0, stream>>>(Fws, memb, mask, out, total4);
}